// NodewiseGraphActor_34918084116568
// MI455X (gfx1250) — compile-verified
//
#include <hip/hip_runtime.h>
#include <hip/hip_bf16.h>
#include <math.h>

typedef __attribute__((ext_vector_type(2))) float v2f;
typedef __attribute__((ext_vector_type(8))) float v8f;

#define Bsz      4096
#define NODES    262144
#define EVS_PER  20
#define Etot     (Bsz * EVS_PER)   // 81920
#define Hdim     128
#define ACT_DIM  200

// ---------------------------------------------------------------------------
// Kernel A: zero the action output (tanh(0) == 0 for all unwritten slots)
// ---------------------------------------------------------------------------
__global__ void zero_out_kernel(float* __restrict__ out, int n) {
    int i = blockIdx.x * blockDim.x + threadIdx.x;
    if (i < n) out[i] = 0.0f;
}

// ---------------------------------------------------------------------------
// Kernel B: context = relu(pooled[:,0,:] @ W_ctx + b_ctx)   [4096,128]
// 256 threads = 8 wave32; block owns 16 rows; wave w owns cols [16w,16w+16)
// ---------------------------------------------------------------------------
__global__ void ctx_kernel(const float* __restrict__ pooled,
                           const float* __restrict__ W_ctx,
                           const float* __restrict__ b_ctx,
                           float* __restrict__ ctx_out) {
    __shared__ float As[16][Hdim + 4];   // +4 pad: row stride mod 64 = 4 -> no bank conflict

    const int tid     = threadIdx.x;
    const int rowbase = blockIdx.x * 16;

    // stage 16x128 A tile
    for (int i = tid; i < 16 * Hdim; i += 256) {
        int r = i >> 7, c = i & 127;
        As[r][c] = pooled[(size_t)(rowbase + r) * Hdim + c];
    }
    __syncthreads();

    const int wave  = tid >> 5;
    const int lane  = tid & 31;
    const int l     = lane & 15;
    const int lhalf = lane >> 4;          // 0: K={k,k+1}, 1: K={k+2,k+3}
    const int ncol  = wave * 16;

    v8f acc = {};
    for (int k = 0; k < Hdim; k += 4) {
        v2f a, b;
        a[0] = As[l][k + 2 * lhalf + 0];
        a[1] = As[l][k + 2 * lhalf + 1];
        b[0] = W_ctx[(size_t)(k + 2 * lhalf + 0) * Hdim + ncol + l];
        b[1] = W_ctx[(size_t)(k + 2 * lhalf + 1) * Hdim + ncol + l];
        acc = __builtin_amdgcn_wmma_f32_16x16x4_f32(
                  false, a, false, b, (short)0, acc, false, false);
    }

    // D layout: VGPR r -> M = r + 8*lhalf, N = l ; fuse bias + ReLU
    const int col = ncol + l;
    const float bias = b_ctx[col];
#pragma unroll
    for (int r = 0; r < 8; ++r) {
        int row = rowbase + r + 8 * lhalf;
        float v = acc[r] + bias;
        ctx_out[(size_t)row * Hdim + col] = v > 0.0f ? v : 0.0f;
    }
}

// ---------------------------------------------------------------------------
// Kernel C: per block of 16 EVs:
//   h   = relu([ev_emb | ctx_seg] @ W1 + b1)        (WMMA, K=256)
//   val = h @ W2 + b2 ; out[seg*200 + mapper] = tanh(val)
// ---------------------------------------------------------------------------
__global__ void ev_kernel(const float* __restrict__ node_emb,
                          const int*   __restrict__ ev_idx,
                          const int*   __restrict__ ev_seg,
                          const int*   __restrict__ act_map,
                          const float* __restrict__ W1,
                          const float* __restrict__ b1,
                          const float* __restrict__ W2,
                          const float* __restrict__ b2,
                          const float* __restrict__ ctx,
                          float* __restrict__ out) {
    __shared__ float As[16][2 * Hdim + 4];   // 16x256 concat tile (padded)
    __shared__ float Hs[16][Hdim + 4];       // 16x128 hidden tile (padded)
    __shared__ int   rowIdx[16];
    __shared__ int   rowSeg[16];

    const int tid    = threadIdx.x;
    const int evbase = blockIdx.x * 16;

    if (tid < 16) {
        rowIdx[tid] = ev_idx[evbase + tid];
        rowSeg[tid] = ev_seg[evbase + tid];
    }
    __syncthreads();

    // stage concat([ev_emb, ctx_per_ev]) : 4096 elems, 16 per thread
    for (int i = tid; i < 16 * 2 * Hdim; i += 256) {
        int r = i >> 8, c = i & 255;
        As[r][c] = (c < Hdim)
                 ? node_emb[(size_t)rowIdx[r] * Hdim + c]
                 : ctx[(size_t)rowSeg[r] * Hdim + (c - Hdim)];
    }
    __syncthreads();

    const int wave  = tid >> 5;
    const int lane  = tid & 31;
    const int l     = lane & 15;
    const int lhalf = lane >> 4;
    const int ncol  = wave * 16;

    v8f acc = {};
    for (int k = 0; k < 2 * Hdim; k += 4) {
        v2f a, b;
        a[0] = As[l][k + 2 * lhalf + 0];
        a[1] = As[l][k + 2 * lhalf + 1];
        b[0] = W1[(size_t)(k + 2 * lhalf + 0) * Hdim + ncol + l];
        b[1] = W1[(size_t)(k + 2 * lhalf + 1) * Hdim + ncol + l];
        acc = __builtin_amdgcn_wmma_f32_16x16x4_f32(
                  false, a, false, b, (short)0, acc, false, false);
    }

    // bias + ReLU into LDS hidden tile
    const int col = ncol + l;
    const float bias = b1[col];
#pragma unroll
    for (int r = 0; r < 8; ++r) {
        int row = r + 8 * lhalf;
        float v = acc[r] + bias;
        Hs[row][col] = v > 0.0f ? v : 0.0f;
    }
    __syncthreads();

    // vals = h @ W2 + b2 ; scatter tanh(val). (seg, mapper) pairs are unique.
    if (tid < 16) {
        float s = b2[0];
#pragma unroll 4
        for (int c = 0; c < Hdim; ++c) s += Hs[tid][c] * W2[c];
        int e = evbase + tid;
        out[(size_t)ev_seg[e] * ACT_DIM + act_map[e]] = tanhf(s);
    }
}

// ---------------------------------------------------------------------------
extern "C" void kernel_launch(void* const* d_in, const int* in_sizes, int n_in,
                              void* d_out, int out_size, void* d_ws, size_t ws_size,
                              hipStream_t stream) {
    const float* pooled   = (const float*)d_in[0];
    const float* node_emb = (const float*)d_in[1];
    const int*   ev_idx   = (const int*)  d_in[2];
    const int*   ev_seg   = (const int*)  d_in[3];
    const int*   act_map  = (const int*)  d_in[4];
    const float* W_ctx    = (const float*)d_in[5];
    const float* b_ctx    = (const float*)d_in[6];
    const float* W1       = (const float*)d_in[7];
    const float* b1       = (const float*)d_in[8];
    const float* W2       = (const float*)d_in[9];
    const float* b2       = (const float*)d_in[10];

    float* out = (float*)d_out;
    float* ctx = (float*)d_ws;               // [B, H] scratch, 2 MB

    // A: zero output
    {
        int n = Bsz * ACT_DIM;
        zero_out_kernel<<<(n + 255) / 256, 256, 0, stream>>>(out, n);
    }
    // B: context projection GEMM + ReLU
    ctx_kernel<<<Bsz / 16, 256, 0, stream>>>(pooled, W_ctx, b_ctx, ctx);
    // C: fused EV MLP + scatter + tanh
    ev_kernel<<<Etot / 16, 256, 0, stream>>>(node_emb, ev_idx, ev_seg, act_map,
                                             W1, b1, W2, b2, ctx, out);
}